// LocalDecoderHyperVoxel_72962904424983
// MI455X (gfx1250) — compile-verified
//
#include <hip/hip_runtime.h>

typedef __attribute__((ext_vector_type(2))) float v2f;
typedef __attribute__((ext_vector_type(8))) float v8f;

constexpr int BB  = 2;
constexpr int NN  = 65536;
constexpr int CD  = 128;
constexpr int HID = 32;
constexpr int TP  = 1217;          // 3*32 + 32 + 32*32 + 32 + 32 + 1
constexpr int GS  = 64;
constexpr int VOX = GS * GS * GS;  // 262144
constexpr int NCOLS_PAD = 77 * 16; // 1232 (zero-padded columns)
constexpr int W3T_STRIDE = 34;     // 32 K-values + bias + pad (keeps b64 alignment)
constexpr int WAVE_LDS = 2048 + 512 + 512 + 512 + 48;  // feats,h1,h2,a,x = 3632 floats

__device__ __forceinline__ float leaky(float v) { return v >= 0.0f ? v : 0.01f * v; }

// ---- prep: transpose weights into fragment-friendly layout in workspace ----
__global__ void prep_kernel(const float* __restrict__ Wh1, const float* __restrict__ Wh2,
                            const float* __restrict__ Wh3, const float* __restrict__ bh3,
                            float* __restrict__ Wh1T, float* __restrict__ Wh2T,
                            float* __restrict__ Wh3T)
{
    const int t = blockIdx.x * blockDim.x + threadIdx.x;
    const int stride = gridDim.x * blockDim.x;
    // Wh3T[c][k] = Wh3[k][c]; slot 32 = bh3[c]; columns >= TP are zero
    for (int idx = t; idx < NCOLS_PAD * W3T_STRIDE; idx += stride) {
        const int c = idx / W3T_STRIDE, k = idx - c * W3T_STRIDE;
        float v = 0.0f;
        if (c < TP) {
            if (k < 32)       v = Wh3[(size_t)k * TP + c];
            else if (k == 32) v = bh3[c];
        }
        Wh3T[idx] = v;
    }
    for (int idx = t; idx < 32 * 128; idx += stride) {   // Wh1T[n][k] = Wh1[k][n]
        const int n = idx >> 7, k = idx & 127;
        Wh1T[idx] = Wh1[k * 32 + n];
    }
    for (int idx = t; idx < 32 * 32; idx += stride) {    // Wh2T[n][k] = Wh2[k][n]
        const int n = idx >> 5, k = idx & 31;
        Wh2T[idx] = Wh2[k * 32 + n];
    }
}

__global__ __launch_bounds__(64) void decoder_kernel(
    const float* __restrict__ pcl_mem, const float* __restrict__ c_plane,
    const float* __restrict__ Wh1T, const float* __restrict__ bh1,
    const float* __restrict__ Wh2T, const float* __restrict__ bh2,
    const float* __restrict__ Wh3T,
    float* __restrict__ out, float* __restrict__ feats_out)
{
    const int lane = threadIdx.x & 31;
    const int wid  = threadIdx.x >> 5;
    const int tile = blockIdx.x * 2 + wid;     // 16-point tile index
    const int g0   = tile * 16;
    const int b    = g0 / NN;
    const int n0   = g0 - b * NN;

    __shared__ float smem[2 * WAVE_LDS];
    float* sF  = smem + wid * WAVE_LDS;   // feats [16][128]
    float* sH1 = sF  + 2048;              // h1    [16][32]
    float* sH2 = sH1 + 512;               // h2    [16][32]
    float* sA  = sH2 + 512;               // leaky(o1) [16][32]
    float* sX  = sA  + 512;               // x = frac-0.5 [16][3]

    // ---------------- Stage A: trilinear grid sample (2 lanes per point) ----
    {
        const int m     = lane >> 1;
        const int chalf = (lane & 1) * 64;
        const float* P  = pcl_mem + (size_t)(b * NN + n0 + m) * 3;
        float vx = P[0], vy = P[1], vz = P[2];

        auto srcidx = [](float v) -> float {
            float g = fminf(fmaxf(2.0f * v * (1.0f / 63.0f) - 1.0f, -2.0f), 2.0f);
            float ix = (g + 1.0f) * 0.5f * 63.0f;
            return fminf(fmaxf(ix, 0.0f), 63.0f);
        };
        float ix = srcidx(vx), iy = srcidx(vy), iz = srcidx(vz);
        float fx = floorf(ix), fy = floorf(iy), fz = floorf(iz);
        float wx = ix - fx, wy = iy - fy, wz = iz - fz;
        int x0 = min(max((int)fx, 0), 63), x1 = min(x0 + 1, 63);
        int y0 = min(max((int)fy, 0), 63), y1 = min(y0 + 1, 63);
        int z0 = min(max((int)fz, 0), 63), z1 = min(z0 + 1, 63);

        const int o000 = (z0*64 + y0)*64 + x0, o001 = (z0*64 + y0)*64 + x1;
        const int o010 = (z0*64 + y1)*64 + x0, o011 = (z0*64 + y1)*64 + x1;
        const int o100 = (z1*64 + y0)*64 + x0, o101 = (z1*64 + y0)*64 + x1;
        const int o110 = (z1*64 + y1)*64 + x0, o111 = (z1*64 + y1)*64 + x1;
        const float ux = 1.0f - wx, uy = 1.0f - wy, uz = 1.0f - wz;
        const float w000 = uz*uy*ux, w001 = uz*uy*wx, w010 = uz*wy*ux, w011 = uz*wy*wx;
        const float w100 = wz*uy*ux, w101 = wz*uy*wx, w110 = wz*wy*ux, w111 = wz*wy*wx;

        const float* Cb = c_plane + (size_t)b * CD * VOX;
        float* fo = feats_out + (size_t)(b * NN + n0 + m) * CD;
        #pragma unroll 4
        for (int c = 0; c < 64; ++c) {
            const float* cc = Cb + (size_t)(chalf + c) * VOX;
            float v = w000*cc[o000] + w001*cc[o001] + w010*cc[o010] + w011*cc[o011]
                    + w100*cc[o100] + w101*cc[o101] + w110*cc[o110] + w111*cc[o111];
            sF[m * CD + chalf + c] = v;
            __builtin_nontemporal_store(v, fo + chalf + c);
        }
        if (lane < 16) {
            const float* P2 = pcl_mem + (size_t)(b * NN + n0 + lane) * 3;
            for (int d = 0; d < 3; ++d) {
                float p = P2[d];
                sX[lane * 3 + d] = p - truncf(p) - 0.5f;
            }
        }
    }
    __syncthreads();

    const int n16 = lane & 15;
    const int hh  = lane >> 4;

    // ---------------- GEMM1: h1 = leaky(feats[16x128] @ Wh1[128x32] + bh1) --
    for (int nt = 0; nt < 2; ++nt) {
        const int n = nt * 16 + n16;
        const float* bt = Wh1T + n * 128;
        v8f acc = {};
        #pragma unroll 4
        for (int ks = 0; ks < 32; ++ks) {
            const int kc = ks * 4 + hh * 2;
            v2f a  = *(const v2f*)(sF + n16 * CD + kc);
            v2f bb = *(const v2f*)(bt + kc);
            acc = __builtin_amdgcn_wmma_f32_16x16x4_f32(false, a, false, bb, (short)0, acc, false, false);
        }
        const float bias = bh1[n];
        #pragma unroll
        for (int r = 0; r < 8; ++r)
            sH1[(hh * 8 + r) * 32 + n] = leaky(acc[r] + bias);
    }
    __syncthreads();

    // ---------------- GEMM2: h2 = leaky(h1[16x32] @ Wh2[32x32] + bh2) -------
    for (int nt = 0; nt < 2; ++nt) {
        const int n = nt * 16 + n16;
        const float* bt = Wh2T + n * 32;
        v8f acc = {};
        #pragma unroll
        for (int ks = 0; ks < 8; ++ks) {
            const int kc = ks * 4 + hh * 2;
            v2f a  = *(const v2f*)(sH1 + n16 * 32 + kc);
            v2f bb = *(const v2f*)(bt + kc);
            acc = __builtin_amdgcn_wmma_f32_16x16x4_f32(false, a, false, bb, (short)0, acc, false, false);
        }
        const float bias = bh2[n];
        #pragma unroll
        for (int r = 0; r < 8; ++r)
            sH2[(hh * 8 + r) * 32 + n] = leaky(acc[r] + bias);
    }
    __syncthreads();

    // ---------------- GEMM3 A fragments held in registers --------------------
    v2f af[8];
    #pragma unroll
    for (int ks = 0; ks < 8; ++ks) {
        const int kc = ks * 4 + hh * 2;
        af[ks] = *(const v2f*)(sH2 + n16 * 32 + kc);
    }

    // one 16-column tile of hw = h2 @ Wh3 + bh3 (bias folded into column 32;
    // padded columns are zero, so no guards needed anywhere)
    auto hwtile = [&](int nt) -> v8f {
        const float* colp = Wh3T + (size_t)(nt * 16 + n16) * W3T_STRIDE;
        const float bias = colp[32];
        v8f acc = {bias, bias, bias, bias, bias, bias, bias, bias};
        #pragma unroll
        for (int ks = 0; ks < 8; ++ks) {
            v2f bb = *(const v2f*)(colp + ks * 4 + hh * 2);
            acc = __builtin_amdgcn_wmma_f32_16x16x4_f32(false, af[ks], false, bb, (short)0, acc, false, false);
        }
        return acc;
    };

    // ---- Phase W1 + b1: o1[p][i] = sum_d x[p][d]*hw[d*32+i] + hw[96+i] ------
    float xm[8][3];
    #pragma unroll
    for (int r = 0; r < 8; ++r)
        for (int d = 0; d < 3; ++d)
            xm[r][d] = sX[(hh * 8 + r) * 3 + d];

    float o1[2][8];
    #pragma unroll
    for (int h = 0; h < 2; ++h) for (int r = 0; r < 8; ++r) o1[h][r] = 0.0f;

    #pragma unroll 1
    for (int d = 0; d < 3; ++d)
        for (int h = 0; h < 2; ++h) {
            v8f Dv = hwtile(d * 2 + h);
            #pragma unroll
            for (int r = 0; r < 8; ++r) o1[h][r] += xm[r][d] * Dv[r];
        }
    for (int h = 0; h < 2; ++h) {
        v8f Dv = hwtile(6 + h);                              // b1 columns
        #pragma unroll
        for (int r = 0; r < 8; ++r)
            sA[(hh * 8 + r) * 32 + h * 16 + n16] = leaky(o1[h][r] + Dv[r]);
    }
    __syncthreads();

    // ---- Phase W2 + b2: o2[p][j] = sum_i a[p][i]*hw[128+i*32+j] + hw[1152+j]
    float o2[2][8];
    #pragma unroll
    for (int h = 0; h < 2; ++h) for (int r = 0; r < 8; ++r) o2[h][r] = 0.0f;

    #pragma unroll 2
    for (int i = 0; i < 32; ++i) {
        float av[8];
        #pragma unroll
        for (int r = 0; r < 8; ++r) av[r] = sA[(hh * 8 + r) * 32 + i];
        for (int h = 0; h < 2; ++h) {
            v8f Dv = hwtile(8 + i * 2 + h);
            #pragma unroll
            for (int r = 0; r < 8; ++r) o2[h][r] += av[r] * Dv[r];
        }
    }
    float a2[2][8];
    for (int h = 0; h < 2; ++h) {
        v8f Dv = hwtile(72 + h);                             // b2 columns
        #pragma unroll
        for (int r = 0; r < 8; ++r) a2[h][r] = leaky(o2[h][r] + Dv[r]);
    }

    // ---- Phase W3 + b3: out[p] = sum_j a2[p][j]*hw[1184+j] + hw[1216] -------
    float part[8];
    #pragma unroll
    for (int r = 0; r < 8; ++r) part[r] = 0.0f;
    for (int h = 0; h < 2; ++h) {
        v8f Dv = hwtile(74 + h);
        #pragma unroll
        for (int r = 0; r < 8; ++r) part[r] += a2[h][r] * Dv[r];
    }
    // reduce over the 16 j-lanes (xor masks < 16 stay within each half-wave)
    #pragma unroll
    for (int msk = 8; msk >= 1; msk >>= 1)
        for (int r = 0; r < 8; ++r)
            part[r] += __shfl_xor(part[r], msk, 32);

    v8f Db3 = hwtile(76);                                    // padded cols are zero
    if (n16 == 0) {
        #pragma unroll
        for (int r = 0; r < 8; ++r)
            out[(size_t)b * NN + n0 + hh * 8 + r] = part[r] + Db3[r];
    }
}

extern "C" void kernel_launch(void* const* d_in, const int* in_sizes, int n_in,
                              void* d_out, int out_size, void* d_ws, size_t ws_size,
                              hipStream_t stream) {
    // inputs: 0 pcl (unused), 1 pcl_mem, 2 c_plane, 3 Wh1, 4 bh1, 5 Wh2, 6 bh2, 7 Wh3, 8 bh3
    const float* pcl_mem = (const float*)d_in[1];
    const float* c_plane = (const float*)d_in[2];
    const float* Wh1 = (const float*)d_in[3];
    const float* bh1 = (const float*)d_in[4];
    const float* Wh2 = (const float*)d_in[5];
    const float* bh2 = (const float*)d_in[6];
    const float* Wh3 = (const float*)d_in[7];
    const float* bh3 = (const float*)d_in[8];

    // workspace layout (all rewritten every call)
    float* Wh3T = (float*)d_ws;                            // 1232*34 floats
    float* Wh1T = Wh3T + NCOLS_PAD * W3T_STRIDE;           // 32*128
    float* Wh2T = Wh1T + 32 * 128;                         // 32*32

    float* out       = (float*)d_out;                      // [B*N]
    float* feats_out = (float*)d_out + (size_t)BB * NN;    // [B*N*128]

    prep_kernel<<<64, 256, 0, stream>>>(Wh1, Wh2, Wh3, bh3, Wh1T, Wh2T, Wh3T);

    const int tiles  = BB * NN / 16;   // 8192 wave-tiles
    const int blocks = tiles / 2;      // 2 waves per block
    decoder_kernel<<<blocks, 64, 0, stream>>>(pcl_mem, c_plane, Wh1T, bh1, Wh2T, bh2,
                                              Wh3T, out, feats_out);
}